// Q_Model_16999480557725
// MI455X (gfx1250) — compile-verified
//
#include <hip/hip_runtime.h>

#define NW   10
#define QDIM 1024

typedef __attribute__((ext_vector_type(2))) float v2f;
typedef __attribute__((ext_vector_type(8))) float v8f;

namespace {

constexpr int pc10(int v) {
  int n = 0;
  for (int i = 0; i < 16; ++i) n += (v >> i) & 1;
  return n;
}

// CX-chain index permutation: psi3[j] = psi2[Q(j)], Q = F_0∘F_1∘...∘F_9,
// F_w(j): if ctrl bit (wire w, position 9-w) set, flip tgt bit (wire (w+1)%10).
constexpr int applyQ(int j) {
  int t = j;
  for (int w = 9; w >= 0; --w) {
    const int ctrl_pos = 9 - w;
    const int tgt_pos  = 9 - ((w + 1) % 10);
    if ((t >> ctrl_pos) & 1) t ^= (1 << tgt_pos);
  }
  return t;
}

struct QTables {
  int ZA[NW][11];  // <Z_w> coefficients vs popcount(Q(j))
  int XB[NW][21];  // <X_w> coefficients vs t = popcount(i)+popcount(i^d_w)
  constexpr QTables() : ZA{}, XB{} {
    for (int j = 0; j < QDIM; ++j) {
      const int k = pc10(applyQ(j));
      for (int w = 0; w < NW; ++w)
        ZA[w][k] += (((j >> (9 - w)) & 1) ? -1 : 1);
    }
    for (int w = 0; w < NW; ++w) {
      const int d = applyQ(1 << (9 - w));  // d_w = Q(e_w), Q is GF(2)-linear
      for (int i = 0; i < QDIM; ++i)
        XB[w][pc10(i) + pc10(i ^ d)] += 1;
    }
  }
};

constexpr QTables hQT{};

} // namespace

__constant__ QTables dQT = hQT;

// Build the padded 16x24 coefficient matrix C[wire][t] into workspace:
//   C[w][t] = cos(ry[w]) * ZA[w][t/2] (t even)  -  sin(ry[w]) * XB[w][t],  t<=20
// Rows 10..15 and cols 21..23 are zero padding.
__global__ void build_coeff_rows(const float* __restrict__ ry,
                                 float* __restrict__ C) {
  const int t = threadIdx.x;  // 0..383
  if (t >= 16 * 24) return;
  const int row = t / 24;
  const int col = t % 24;
  float v = 0.0f;
  if (row < NW && col <= 20) {
    float cw, sw;
    __sincosf(ry[row], &sw, &cw);
    v = -sw * (float)dQT.XB[row][col];
    if ((col & 1) == 0) v += cw * (float)dQT.ZA[row][col >> 1];
  }
  C[t] = v;
}

__global__ __launch_bounds__(256) void q_expz_wmma(const float* __restrict__ x,
                                                   const float* __restrict__ Cmat,
                                                   float* __restrict__ out,
                                                   int B) {
  const int lane = threadIdx.x & 31;
  const int wave = threadIdx.x >> 5;
  const int hi   = lane >> 4;   // half-wave: selects K pair (0,1) vs (2,3) in each 4-chunk
  const int sub  = lane & 15;
  const bool hib = (hi != 0);

  const int tile = blockIdx.x * (blockDim.x >> 5) + wave;
  const int base = tile << 4;                 // 16 samples per wave tile
  if (base >= B) return;                      // wave-uniform: EXEC stays full for WMMA

  // ---- A operand: monomials c^(20-t) * s^t for sample base+sub (pure scalars, no arrays)
  int sidx = base + sub;
  if (sidx >= B) sidx = B - 1;
  float c1, s1;
  __sincosf(0.5f * x[sidx], &s1, &c1);

#define PWSTEP(n, p) const float c##n = c##p * c1; const float s##n = s##p * s1;
  PWSTEP(2, 1)  PWSTEP(3, 2)  PWSTEP(4, 3)  PWSTEP(5, 4)  PWSTEP(6, 5)
  PWSTEP(7, 6)  PWSTEP(8, 7)  PWSTEP(9, 8)  PWSTEP(10, 9) PWSTEP(11, 10)
  PWSTEP(12, 11) PWSTEP(13, 12) PWSTEP(14, 13) PWSTEP(15, 14) PWSTEP(16, 15)
  PWSTEP(17, 16) PWSTEP(18, 17) PWSTEP(19, 18) PWSTEP(20, 19)
#undef PWSTEP

  const float mono0 = c20;
#define MONO(t, u) const float mono##t = c##u * s##t;
  MONO(1, 19)  MONO(2, 18)  MONO(3, 17)  MONO(4, 16)  MONO(5, 15)
  MONO(6, 14)  MONO(7, 13)  MONO(8, 12)  MONO(9, 11)  MONO(10, 10)
  MONO(11, 9)  MONO(12, 8)  MONO(13, 7)  MONO(14, 6)  MONO(15, 5)
  MONO(16, 4)  MONO(17, 3)  MONO(18, 2)  MONO(19, 1)
#undef MONO
  const float mono20 = s20;
  const float mono21 = 0.0f;
  const float mono22 = 0.0f;
  const float mono23 = 0.0f;

  // ---- B operand: lane holds wire column N=sub; K pair per chunk is a contiguous
  // 8-byte load at C[sub][4k + 2*hi] (VGPR0 <- K=4k+2hi, VGPR1 <- K=4k+2hi+1).
  const float* crow = Cmat + sub * 24 + (hi << 1);

  // D[16 samples x 16 wires] = A[16x24] * B[24x16] via six 16x16x4 f32 WMMAs.
  v8f acc = {};
#define KSTEP(k, T0, T1, T2, T3)                                                   \
  {                                                                                \
    v2f a = { hib ? mono##T2 : mono##T0, hib ? mono##T3 : mono##T1 };              \
    v2f b = *(const v2f*)(crow + 4 * (k));                                         \
    acc = __builtin_amdgcn_wmma_f32_16x16x4_f32(false, a, false, b, (short)0, acc, \
                                                false, false);                     \
  }
  KSTEP(0, 0, 1, 2, 3)
  KSTEP(1, 4, 5, 6, 7)
  KSTEP(2, 8, 9, 10, 11)
  KSTEP(3, 12, 13, 14, 15)
  KSTEP(4, 16, 17, 18, 19)
  KSTEP(5, 20, 21, 22, 23)
#undef KSTEP

  // ---- store: lane holds wire N=sub; VGPR r holds sample row M = r + 8*hi
  if (sub < NW) {
#pragma unroll
    for (int r = 0; r < 8; ++r) {
      const int m = base + r + (hi << 3);
      if (m < B) out[m * NW + sub] = acc[r];
    }
  }
}

extern "C" void kernel_launch(void* const* d_in, const int* in_sizes, int n_in,
                              void* d_out, int out_size, void* d_ws, size_t ws_size,
                              hipStream_t stream) {
  (void)n_in; (void)ws_size; (void)out_size;
  const float* x  = (const float*)d_in[0];   // 65536 angles
  const float* ry = (const float*)d_in[1];   // 10 fixed RY params
  float* out  = (float*)d_out;               // [B, 10] f32
  float* Cmat = (float*)d_ws;                // 16x24 padded coefficient matrix

  const int B = in_sizes[0];

  build_coeff_rows<<<1, 384, 0, stream>>>(ry, Cmat);

  const int tiles = (B + 15) / 16;           // 16 samples per wave
  const int waves_per_block = 8;             // 256 threads, wave32
  const int blocks = (tiles + waves_per_block - 1) / waves_per_block;
  q_expz_wmma<<<blocks, 256, 0, stream>>>(x, Cmat, out, B);
}